// CRF_61589831025014
// MI455X (gfx1250) — compile-verified
//
#include <hip/hip_runtime.h>

typedef __attribute__((ext_vector_type(2))) float v2f;
typedef __attribute__((ext_vector_type(8))) float v8f;

#define BDIM 64
#define TDIM 512
#define DDIM 1024
#define KDIM 25

// ------------------------------------------------------------------
// Kernel 1: scores = (inputs @ W + b) * mask   (memory-bound GEMM)
// M=32768 rows, K=1024, N=25 (padded to 32). FP32 WMMA 16x16x4.
// Each wave: one 16-row tile x two 16-col tiles. W staged via LDS
// in 128-row chunks, padded stride 33 to dodge bank conflicts.
// ------------------------------------------------------------------
#define CHUNK 128
#define WSTRIDE 33

__global__ __launch_bounds__(256) void k_gemm(
    const float* __restrict__ inputs, const int* __restrict__ mask,
    const float* __restrict__ W, const float* __restrict__ bias,
    float* __restrict__ scores)
{
  __shared__ float Wp[CHUNK * WSTRIDE];          // 16.9 KB
  const int tid  = threadIdx.x;
  const int lane = tid & 31;
  const int wave = tid >> 5;
  const int m = lane & 15;       // row-in-tile (A) / col-in-tile (B,C)
  const int h = lane >> 4;       // lane half selects K pair / M half
  const int rowBase = (blockIdx.x * 8 + wave) * 16;
  const float* arow = inputs + (size_t)(rowBase + m) * DDIM + 2 * h;

  v8f c0 = {}; v8f c1 = {};
  for (int kb = 0; kb < DDIM; kb += CHUNK) {
    __syncthreads();
    for (int idx = tid; idx < CHUNK * 32; idx += 256) {
      int r = idx >> 5, c = idx & 31;
      Wp[r * WSTRIDE + c] = (c < KDIM) ? W[(kb + r) * KDIM + c] : 0.0f;
    }
    __syncthreads();
#pragma unroll 4
    for (int k = 0; k < CHUNK; k += 4) {
      // A 16x4 f32 layout: lanes 0-15 hold K={k,k+1}, lanes 16-31 K={k+2,k+3}
      v2f a = *(const v2f*)(arow + kb + k);
      // B 4x16 layout mirrors: lane half selects K row pair, m = column
      int r0 = (k + 2 * h) * WSTRIDE;
      v2f b0, b1;
      b0[0] = Wp[r0 + m];            b0[1] = Wp[r0 + WSTRIDE + m];
      b1[0] = Wp[r0 + 16 + m];       b1[1] = Wp[r0 + WSTRIDE + 16 + m];
      c0 = __builtin_amdgcn_wmma_f32_16x16x4_f32(false, a, false, b0,
                                                 (short)0, c0, false, false);
      c1 = __builtin_amdgcn_wmma_f32_16x16x4_f32(false, a, false, b1,
                                                 (short)0, c1, false, false);
    }
  }

  // C/D layout: VGPR r -> row (r + 8*h), col = m (+16 for tile 1)
  float bias0 = bias[m];
  float bias1 = (m < KDIM - 16) ? bias[16 + m] : 0.0f;
#pragma unroll
  for (int r = 0; r < 8; ++r) {
    int row = rowBase + r + 8 * h;
    float mk = (float)mask[row];
    scores[(size_t)row * KDIM + m] = (c0[r] + bias0) * mk;
    if (m < KDIM - 16)
      scores[(size_t)row * KDIM + 16 + m] = (c1[r] + bias1) * mk;
  }
}

// ------------------------------------------------------------------
// Wave32 reductions
// ------------------------------------------------------------------
__device__ inline float wave_sum(float v) {
  for (int o = 16; o > 0; o >>= 1) v += __shfl_xor(v, o, 32);
  return v;
}
__device__ inline float wave_max(float v) {
  for (int o = 16; o > 0; o >>= 1) v = fmaxf(v, __shfl_xor(v, o, 32));
  return v;
}

// ------------------------------------------------------------------
// Kernel 2: fused CRF forward (log-denominator) + Viterbi + joint
// score. One wave per batch element; lane j = state j (j<25).
//
// Scan-step optimization: with E_ij = exp(trans_ij) precomputed once,
//   logsumexp_i(alpha_i + trans_ij) = m + log(sum_i exp(alpha_i-m)*E_ij)
// where m = max_i alpha_i (wave max). Each step then costs one
// v_exp + one v_log per lane plus FMAs, instead of 25 v_exp.
// Viterbi argmax uses a depth-5 tournament tree (ties keep the left slot).
// Backpointers live in LDS (16 KB) for the sequential backtrack.
// ------------------------------------------------------------------
__global__ __launch_bounds__(32) void k_crf(
    const float* __restrict__ scores, const int* __restrict__ mask,
    const int* __restrict__ labels, const float* __restrict__ trans,
    const float* __restrict__ start_t, const float* __restrict__ end_t,
    float* __restrict__ tags_out, float* __restrict__ llh)
{
  __shared__ unsigned char bp[TDIM * 32];
  __shared__ float red[32];
  const int b = blockIdx.x;
  const int j = threadIdx.x;
  const bool act = (j < KDIM);
  const float NEG = -1e30f;

  float tc[KDIM];    // trans[:, j]
  float Ec[KDIM];    // exp(trans[:, j])
#pragma unroll
  for (int i = 0; i < KDIM; ++i) {
    tc[i] = act ? trans[i * KDIM + j] : NEG;
    Ec[i] = act ? __expf(tc[i]) : 0.0f;
  }

  const float* sc = scores + (size_t)b * TDIM * KDIM;
  const int* mb = mask + b * TDIM;
  const int* lb = labels + b * TDIM;

  float a_lse = act ? (start_t[j] + sc[j]) : NEG;
  float a_vit = a_lse;

  for (int t = 1; t < TDIM; ++t) {
    float logit = act ? sc[t * KDIM + j] : 0.0f;
    int mt = mb[t];

    // ---- logsumexp recurrence via single exp/log per step ----
    float m = wave_max(a_lse);               // lanes >=25 hold NEG
    float p = act ? __expf(a_lse - m) : 0.0f;

    float s0 = 0, s1 = 0, s2 = 0, s3 = 0;
    float uv[32]; int ui[32];
#pragma unroll
    for (int i = 0; i < KDIM; ++i) {
      float pi = __shfl(p, i, 32);
      float av = __shfl(a_vit, i, 32);
      if ((i & 3) == 0)      s0 = fmaf(pi, Ec[i], s0);
      else if ((i & 3) == 1) s1 = fmaf(pi, Ec[i], s1);
      else if ((i & 3) == 2) s2 = fmaf(pi, Ec[i], s2);
      else                   s3 = fmaf(pi, Ec[i], s3);
      uv[i] = av + tc[i];
      ui[i] = i;
    }
#pragma unroll
    for (int i = KDIM; i < 32; ++i) { uv[i] = NEG; ui[i] = i; }

    // ---- tournament argmax (strict > keeps slot i on ties; not always first max) ----
#pragma unroll
    for (int st = 16; st >= 1; st >>= 1) {
#pragma unroll
      for (int i = 0; i < 16; ++i) {
        if (i < st) {
          bool g = uv[i + st] > uv[i];
          uv[i] = g ? uv[i + st] : uv[i];
          ui[i] = g ? ui[i + st] : ui[i];
        }
      }
    }

    float nl = m + __logf((s0 + s1) + (s2 + s3)) + logit;
    float nv = uv[0] + logit;
    int arg = ui[0];
    if (mt && act) { a_lse = nl; a_vit = nv; }
    bp[t * 32 + j] = (unsigned char)(mt ? arg : j);
  }

  // ---- log denominator: logsumexp over states of (alpha + end_t) ----
  float fv = act ? (a_lse + end_t[j]) : NEG;
  float fm = wave_max(fv);
  float fs = wave_sum(act ? __expf(fv - fm) : 0.0f);
  float logden = fm + __logf(fs);

  // ---- joint (gold-path) score, lanes strided over t ----
  float part = 0.0f;
  int cnt = 0;
  for (int t = j; t < TDIM; t += 32) {
    float mkf = (float)mb[t];
    int lab = lb[t];
    if (t < TDIM - 1) part += sc[t * KDIM + lab] * mkf;
    if (t >= 1)       part += trans[lb[t - 1] * KDIM + lab] * mkf;
    cnt += mb[t];
  }
  float total = wave_sum(part);
  int len = (int)(wave_sum((float)cnt) + 0.5f);
  int ltag = lb[len - 1];
  float joint = total + start_t[lb[0]] + end_t[ltag]
              + sc[(TDIM - 1) * KDIM + ltag] * (float)mb[TDIM - 1];

  // ---- Viterbi terminal state + backtrack (lane 0) ----
  red[j] = act ? (a_vit + end_t[j]) : NEG;
  __syncthreads();
  if (j == 0) {
    llh[b] = joint - logden;
    float bestv = red[0]; int cur = 0;
    for (int i = 1; i < KDIM; ++i)
      if (red[i] > bestv) { bestv = red[i]; cur = i; }
    float* tg = tags_out + b * TDIM;
    tg[TDIM - 1] = (float)cur;
    for (int t = TDIM - 1; t >= 1; --t) {
      cur = bp[t * 32 + cur];
      tg[t - 1] = (float)cur;
    }
  }
}

// ------------------------------------------------------------------
// Kernel 3: loss = -mean(llh)
// ------------------------------------------------------------------
__global__ __launch_bounds__(64) void k_loss(const float* __restrict__ llh,
                                             float* __restrict__ out)
{
  __shared__ float s[64];
  s[threadIdx.x] = llh[threadIdx.x];
  __syncthreads();
  if (threadIdx.x == 0) {
    float t = 0.0f;
    for (int i = 0; i < 64; ++i) t += s[i];
    out[0] = -t / (float)BDIM;
  }
}

extern "C" void kernel_launch(void* const* d_in, const int* in_sizes, int n_in,
                              void* d_out, int out_size, void* d_ws, size_t ws_size,
                              hipStream_t stream) {
  const float* inputs  = (const float*)d_in[0];
  const int*   mask    = (const int*)d_in[1];
  const int*   labels  = (const int*)d_in[2];
  const float* W       = (const float*)d_in[3];
  const float* bias    = (const float*)d_in[4];
  const float* trans   = (const float*)d_in[5];
  const float* start_t = (const float*)d_in[6];
  const float* end_t   = (const float*)d_in[7];

  float* scores = (float*)d_out;                       // B*T*K floats
  float* tags   = scores + (size_t)BDIM * TDIM * KDIM; // B*T (as float)
  float* loss   = tags + (size_t)BDIM * TDIM;          // 1
  float* llh    = (float*)d_ws;                        // 64 floats scratch

  // 2048 row tiles of 16, 8 waves per block -> 256 blocks
  k_gemm<<<256, 256, 0, stream>>>(inputs, mask, W, bias, scores);
  k_crf<<<BDIM, 32, 0, stream>>>(scores, mask, labels, trans, start_t, end_t,
                                 tags, llh);
  k_loss<<<1, 64, 0, stream>>>(llh, loss);
}